// GRAND_25451976196187
// MI455X (gfx1250) — compile-verified
//
#include <hip/hip_runtime.h>
#include <math.h>

#define TAU 0.25f

typedef __attribute__((ext_vector_type(2))) float v2f;
typedef __attribute__((ext_vector_type(8))) float v8f;

// ---------------------------------------------------------------------------
// D(MxNcols) = A(MxK) @ B(KxNcols) + bias, f32 via V_WMMA_F32_16X16X4_F32.
// Block = 128 threads = 4 waves; each wave owns one 16-row strip and computes
// NT 16x16 tiles across the full (padded) column range. B is staged once per
// block into LDS, transposed (Bt[n][k], stride K+2, zero-padded cols).
// Inner loop: A fragment prefetched one k-step ahead (global b64), all NT B
// fragments loaded into separate registers (batched ds loads, single wait),
// then NT WMMAs back-to-back. No divergent guards inside the loop.
// Lane layouts per CDNA5 ISA 7.12.2:
//   A 16x4 : lane L -> row m=L&15, k-pair base 2*(L>>4)
//   B 4x16 : lane L -> col n=L&15, k-pair base 2*(L>>4)
//   C 16x16: VGPR i -> row i + 8*(L>>4), col L&15
// ---------------------------------------------------------------------------
template <int NT>
__global__ void __launch_bounds__(128)
gemm_bias_wmma(const float* __restrict__ A, const float* __restrict__ B,
               const float* __restrict__ bias, float* __restrict__ D,
               int M, int K, int Ncols) {
  extern __shared__ float Bt[];          // [NT*16][K+2]
  const int tid = threadIdx.x;
  const int Np  = NT * 16;
  const int Kp  = K + 2;                 // even stride: b64 alignment + bank spread

  // ---- cooperative B load: global (coalesced, n fastest) -> LDS transposed
  for (int idx = tid; idx < Np * K; idx += blockDim.x) {
    int k = idx / Np;
    int n = idx - k * Np;
    float v = (n < Ncols) ? B[k * Ncols + n] : 0.0f;
    Bt[n * Kp + k] = v;
  }
  __syncthreads();

  const int wave = tid >> 5;
  const int lane = tid & 31;
  const int mt   = blockIdx.x * (blockDim.x >> 5) + wave;
  if (mt * 16 >= M) return;              // uniform per wave

  const int nl   = lane & 15;
  const int half = lane >> 4;
  const float* __restrict__ Arow = A + (size_t)(mt * 16 + nl) * K + half * 2;
  const float* __restrict__ Brow = Bt + nl * Kp + half * 2;   // + nt*16*Kp per tile

  v8f acc[NT];
#pragma unroll
  for (int nt = 0; nt < NT; ++nt) acc[nt] = (v8f){};

  v2f a_cur = *(const v2f*)(Arow);       // prefetch k = 0
#pragma unroll 4
  for (int k = 0; k < K - 4; k += 4) {
    v2f a_next = *(const v2f*)(Arow + k + 4);            // prefetch next step
    v2f bf[NT];
#pragma unroll
    for (int nt = 0; nt < NT; ++nt)                      // batched ds loads
      bf[nt] = *(const v2f*)(Brow + nt * 16 * Kp + k);
#pragma unroll
    for (int nt = 0; nt < NT; ++nt)                      // back-to-back WMMAs
      acc[nt] = __builtin_amdgcn_wmma_f32_16x16x4_f32(false, a_cur, false, bf[nt],
                                                      (short)0, acc[nt],
                                                      false, false);
    a_cur = a_next;
  }
  {                                                      // epilogue k-step
    const int k = K - 4;
    v2f bf[NT];
#pragma unroll
    for (int nt = 0; nt < NT; ++nt)
      bf[nt] = *(const v2f*)(Brow + nt * 16 * Kp + k);
#pragma unroll
    for (int nt = 0; nt < NT; ++nt)
      acc[nt] = __builtin_amdgcn_wmma_f32_16x16x4_f32(false, a_cur, false, bf[nt],
                                                      (short)0, acc[nt],
                                                      false, false);
  }

  // ---- epilogue: bias + guarded store (only place the pad guard exists)
#pragma unroll
  for (int nt = 0; nt < NT; ++nt) {
    const int n = nt * 16 + nl;
    if (n < Ncols) {
      float bv = bias[n];
#pragma unroll
      for (int i = 0; i < 8; ++i) {
        int m = mt * 16 + i + half * 8;
        D[(size_t)m * Ncols + n] = acc[nt][i] + bv;
      }
    }
  }
}

// ---------------------------------------------------------------------------
// Per edge: w_e = exp(dot(kx[u], qx[v]) / H); accumulate rowsum[u], count[u].
// ---------------------------------------------------------------------------
__global__ void edge_score_kernel(const int* __restrict__ edges,
                                  const float* __restrict__ kx,
                                  const float* __restrict__ qx,
                                  float* __restrict__ w,
                                  float* __restrict__ rowsum,
                                  int* __restrict__ count,
                                  int E, int H) {
  int e = blockIdx.x * blockDim.x + threadIdx.x;
  if (e >= E) return;
  int u = edges[2 * e + 0];
  int v = edges[2 * e + 1];
  const float4* ku = (const float4*)(kx + (long)u * H);
  const float4* qv = (const float4*)(qx + (long)v * H);
  float s = 0.0f;
  for (int i = 0; i < H / 4; ++i) {
    float4 a = ku[i], b = qv[i];
    s += a.x * b.x + a.y * b.y + a.z * b.z + a.w * b.w;
  }
  float wv = expf(s / (float)H);
  w[e] = wv;
  atomicAdd(&rowsum[u], wv);
  atomicAdd(&count[u], 1);
}

// Single-block exclusive prefix scan over per-row counts (N % 256 == 0).
__global__ void scan_rows(const int* __restrict__ count,
                          int* __restrict__ rowstart, int N) {
  __shared__ int partial[256];
  __shared__ int base[256];
  int t = threadIdx.x;
  int chunk = N / 256;
  int s = 0;
  for (int i = 0; i < chunk; ++i) s += count[t * chunk + i];
  partial[t] = s;
  __syncthreads();
  if (t == 0) {
    int run = 0;
    for (int i = 0; i < 256; ++i) { base[i] = run; run += partial[i]; }
    rowstart[N] = run;
  }
  __syncthreads();
  int off = base[t];
  for (int i = 0; i < chunk; ++i) {
    rowstart[t * chunk + i] = off;
    off += count[t * chunk + i];
  }
}

// Fill CSR: col[] and normalized weight tau * w_e / rowsum[u].
__global__ void csr_fill_kernel(const int* __restrict__ edges,
                                const float* __restrict__ w,
                                const float* __restrict__ rowsum,
                                const int* __restrict__ rowstart,
                                int* __restrict__ cursor,
                                int* __restrict__ col,
                                float* __restrict__ wcsr, int E) {
  int e = blockIdx.x * blockDim.x + threadIdx.x;
  if (e >= E) return;
  int u = edges[2 * e + 0];
  int v = edges[2 * e + 1];
  int pos = rowstart[u] + atomicAdd(&cursor[u], 1);
  col[pos] = v;
  float rs = rowsum[u];
  rs = (rs == 0.0f) ? 1.0f : rs;
  wcsr[pos] = TAU * w[e] / rs;
}

// One diffusion step: y[u] = (1-tau)*x[u] + sum_j wcsr[j] * x[col[j]].
// One wave per node; 32 lanes x float2 cover H=64 features.
__global__ void __launch_bounds__(256)
diffuse_kernel(const float* __restrict__ xin, float* __restrict__ xout,
               const int* __restrict__ rowstart, const int* __restrict__ col,
               const float* __restrict__ wcsr, int N) {
  int node = blockIdx.x * (blockDim.x >> 5) + (threadIdx.x >> 5);
  if (node >= N) return;
  int lane = threadIdx.x & 31;
  const float2* xi = (const float2*)xin;        // 32 float2 per 64-wide row
  float2 acc = xi[(long)node * 32 + lane];
  acc.x *= (1.0f - TAU);
  acc.y *= (1.0f - TAU);
  int jb = rowstart[node];
  int je = rowstart[node + 1];
  for (int j = jb; j < je; ++j) {
    int   c  = col[j];
    float wv = wcsr[j];
    float2 xv = xi[(long)c * 32 + lane];
    acc.x += wv * xv.x;
    acc.y += wv * xv.y;
  }
  ((float2*)xout)[(long)node * 32 + lane] = acc;
}

// ---------------------------------------------------------------------------
extern "C" void kernel_launch(void* const* d_in, const int* in_sizes, int n_in,
                              void* d_out, int out_size, void* d_ws, size_t ws_size,
                              hipStream_t stream) {
  const float* x_in  = (const float*)d_in[0];
  const float* enc_w = (const float*)d_in[1];
  const float* enc_b = (const float*)d_in[2];
  const float* wk_w  = (const float*)d_in[3];
  const float* wk_b  = (const float*)d_in[4];
  const float* wq_w  = (const float*)d_in[5];
  const float* wq_b  = (const float*)d_in[6];
  const float* dec_w = (const float*)d_in[7];
  const float* dec_b = (const float*)d_in[8];
  const int*   edges = (const int*)d_in[9];

  const int H    = in_sizes[2];            // 64
  const int D_IN = in_sizes[1] / H;        // 128
  const int N    = in_sizes[0] / D_IN;     // 8192
  const int C    = in_sizes[8];            // 40
  const int E    = in_sizes[9] / 2;        // 262144
  const int STEPS = 16;                    // ceil(T / TAU), T = 4

  // ---- workspace layout (bytes) ----
  char* ws = (char*)d_ws;
  size_t off = 0;
  const size_t NH = (size_t)N * H * sizeof(float);
  float* x0   = (float*)(ws + off); off += NH;
  float* kx   = (float*)(ws + off); off += NH;
  float* qx   = (float*)(ws + off); off += NH;
  float* xA   = (float*)(ws + off); off += NH;
  float* xB   = (float*)(ws + off); off += NH;
  float* wbuf = (float*)(ws + off); off += (size_t)E * sizeof(float);
  float* wcsr = (float*)(ws + off); off += (size_t)E * sizeof(float);
  int*   colb = (int*)  (ws + off); off += (size_t)E * sizeof(int);
  float* rsum = (float*)(ws + off); off += (size_t)N * sizeof(float);
  int*   cnt  = (int*)  (ws + off); off += (size_t)N * sizeof(int);
  int*   cur  = (int*)  (ws + off); off += (size_t)N * sizeof(int);
  int*   rstr = (int*)  (ws + off); off += (size_t)(N + 1) * sizeof(int);

  // Zero the accumulators each call (rsum, cnt, cur are contiguous).
  hipMemsetAsync(rsum, 0, (size_t)3 * N * sizeof(int), stream);

  // ---- encoder / K / Q GEMMs via f32 WMMA ----
  {
    int grid = (N / 16) / 4;                            // 4 waves (m-tiles) / block
    size_t lds_enc = (size_t)(4 * 16) * (D_IN + 2) * sizeof(float);
    size_t lds_h   = (size_t)(4 * 16) * (H + 2) * sizeof(float);
    gemm_bias_wmma<4><<<grid, 128, lds_enc, stream>>>(x_in, enc_w, enc_b, x0, N, D_IN, H);
    gemm_bias_wmma<4><<<grid, 128, lds_h,   stream>>>(x0, wk_w, wk_b, kx, N, H, H);
    gemm_bias_wmma<4><<<grid, 128, lds_h,   stream>>>(x0, wq_w, wq_b, qx, N, H, H);
  }

  // ---- edge attention weights + CSR build ----
  edge_score_kernel<<<(E + 255) / 256, 256, 0, stream>>>(edges, kx, qx, wbuf, rsum, cnt, E, H);
  scan_rows<<<1, 256, 0, stream>>>(cnt, rstr, N);
  csr_fill_kernel<<<(E + 255) / 256, 256, 0, stream>>>(edges, wbuf, rsum, rstr, cur, colb, wcsr, E);

  // ---- 16 diffusion steps, ping-pong ----
  const float* xcur = x0;
  {
    int nodesPerBlock = 8;                              // 256 threads = 8 waves
    int grid = (N + nodesPerBlock - 1) / nodesPerBlock;
    for (int s = 0; s < STEPS; ++s) {
      float* xnext = (s & 1) ? xB : xA;
      diffuse_kernel<<<grid, 256, 0, stream>>>(xcur, xnext, rstr, colb, wcsr, N);
      xcur = xnext;
    }
  }

  // ---- decode GEMM (C = 40 -> padded to 48 cols, guard only at store) ----
  {
    int grid = (N / 16) / 4;
    size_t lds_dec = (size_t)(3 * 16) * (H + 2) * sizeof(float);
    gemm_bias_wmma<3><<<grid, 128, lds_dec, stream>>>(xcur, dec_w, dec_b, (float*)d_out, N, H, C);
  }
}